// GCN_16260746182861
// MI455X (gfx1250) — compile-verified
//
#include <hip/hip_runtime.h>
#include <hip/hip_bf16.h>
#include <stdint.h>

// GCN 2-layer forward for MI455X (gfx1250, wave32).
// fp32 WMMA (v_wmma_f32_16x16x4_f32) for the dense feature transforms,
// wave-per-edge coalesced gather + global_atomic_add_f32 scatter for the
// normalized aggregation. All hot data (features 25.6MB, accumulators,
// edge index 25.6MB) is resident in the 192MB L2.

#define N_NODES 100000
#define N_EDGES 1600000
#define DFEAT   64

typedef __attribute__((ext_vector_type(2))) float v2f;
typedef __attribute__((ext_vector_type(8))) float v8f;

// ---------------------------------------------------------------------------
// Degree / normalization kernels
// ---------------------------------------------------------------------------
__global__ void gcn_deg_init(float* __restrict__ deg, int n) {
    int i = blockIdx.x * blockDim.x + threadIdx.x;
    if (i < n) deg[i] = 1.0f;  // self-loop contributes 1
}

__global__ void gcn_deg_count(const long long* __restrict__ cols,
                              float* __restrict__ deg, int e) {
    int i = blockIdx.x * blockDim.x + threadIdx.x;
    if (i < e) atomicAdd(&deg[(int)cols[i]], 1.0f);
}

__global__ void gcn_deg_rsqrt(float* __restrict__ deg, int n) {
    int i = blockIdx.x * blockDim.x + threadIdx.x;
    if (i < n) deg[i] = rsqrtf(deg[i]);
}

// ---------------------------------------------------------------------------
// Bias pre-init of the scatter accumulator: out = b (then segment-sum adds)
// ---------------------------------------------------------------------------
__global__ void gcn_bias_init(float* __restrict__ dst,
                              const float* __restrict__ bias, int n_elems) {
    int i = blockIdx.x * blockDim.x + threadIdx.x;
    if (i < n_elems) dst[i] = bias[i & (DFEAT - 1)];
}

// ---------------------------------------------------------------------------
// Dense transform: C[N,64] = act(A[N,64]) @ B[64,64] via V_WMMA_F32_16X16X4_F32
// One wave computes one 16x16 tile of C. 16 WMMA steps over K=64.
// N is an exact multiple of 16 (100000 = 6250*16): no tail handling.
// ISA layouts (CDNA5 ISA §7.12.2):
//   A 16x4 f32 : lane m (0..15) row M=m holds K={k0,k0+1}; lanes 16..31 K+=2
//   B 4x16 f32 : vgpr j holds row K=k0+j (lanes 0-15) / K=k0+2+j (lanes 16-31)
//   C/D 16x16  : vgpr i -> M=i (lanes 0-15), M=i+8 (lanes 16-31), N=lane%16
// RELU is a compile-time template parameter so the non-relu instantiation is
// pure load+wmma and the relu instantiation needs no per-element cndmask.
// ---------------------------------------------------------------------------
template <bool RELU>
__global__ void gcn_gemm_wmma(const float* __restrict__ A,
                              const float* __restrict__ B,
                              float* __restrict__ C) {
    const int wave   = (blockIdx.x * blockDim.x + threadIdx.x) >> 5;
    const int lane   = threadIdx.x & 31;
    const int half   = lane >> 4;      // 0: lanes 0-15, 1: lanes 16-31
    const int m      = lane & 15;

    const int row_tile = wave >> 2;          // 6250 row tiles
    const int n0       = (wave & 3) * 16;    // 4 column tiles (DFEAT=64)
    const int row0     = row_tile * 16;

    v8f acc = {};

    #pragma unroll
    for (int k0 = 0; k0 < DFEAT; k0 += 4) {
        const int ka = k0 + 2 * half;  // this lane's K pair start
        // A operand: contiguous float2 from row (row0+m)
        v2f a;
        {
            const float* ap = A + (size_t)(row0 + m) * DFEAT + ka;
            float ax = ap[0];
            float ay = ap[1];
            if (RELU) { ax = fmaxf(ax, 0.0f); ay = fmaxf(ay, 0.0f); }
            a.x = ax; a.y = ay;
        }
        // B operand: rows ka, ka+1 of B, column n0+m
        v2f b;
        {
            const float* bp = B + (size_t)ka * DFEAT + n0 + m;
            b.x = bp[0];
            b.y = bp[DFEAT];
        }
        acc = __builtin_amdgcn_wmma_f32_16x16x4_f32(
            /*neg_a=*/false, a, /*neg_b=*/false, b,
            /*c_mod=*/(short)0, acc, /*reuse_a=*/false, /*reuse_b=*/false);
    }

    // Store D tile
    #pragma unroll
    for (int i = 0; i < 8; ++i) {
        C[(size_t)(row0 + i + 8 * half) * DFEAT + n0 + m] = acc[i];
    }
}

// ---------------------------------------------------------------------------
// Normalized gather-scatter: one wave per (edge or self-loop).
// Lane l handles features {2l, 2l+1}: coalesced 256B gather per edge,
// coalesced f32 atomic adds into the L2-resident accumulator.
// Virtual edges e >= n_edges are the self-loops (row = col = e - n_edges).
// ---------------------------------------------------------------------------
__global__ void gcn_scatter(const float* __restrict__ src,      // [N,64] transformed
                            const long long* __restrict__ rows,
                            const long long* __restrict__ cols,
                            const float* __restrict__ dis,      // deg^{-1/2}
                            float* __restrict__ dst,            // [N,64] pre-init w/ bias
                            int n_edges, int n_nodes) {
    const int wave = (blockIdx.x * blockDim.x + threadIdx.x) >> 5;
    const int lane = threadIdx.x & 31;
    const int total = n_edges + n_nodes;
    if (wave >= total) return;

    int r, c;
    float norm;
    if (wave < n_edges) {
        r = (int)rows[wave];
        c = (int)cols[wave];
        norm = dis[r] * dis[c];
    } else {
        r = c = wave - n_edges;
        float d = dis[r];
        norm = d * d;
    }

    const float2* sp = (const float2*)(src + (size_t)r * DFEAT);
    float2 v = sp[lane];
    float* dp = dst + (size_t)c * DFEAT + 2 * lane;
    atomicAdd(&dp[0], v.x * norm);
    atomicAdd(&dp[1], v.y * norm);
}

// ---------------------------------------------------------------------------
// Launcher
// ---------------------------------------------------------------------------
extern "C" void kernel_launch(void* const* d_in, const int* in_sizes, int n_in,
                              void* d_out, int out_size, void* d_ws, size_t ws_size,
                              hipStream_t stream) {
    const float*     x    = (const float*)d_in[0];
    const long long* ei   = (const long long*)d_in[1];   // int64 [2, E]
    const float*     W1   = (const float*)d_in[2];
    const float*     b1   = (const float*)d_in[3];
    const float*     W2   = (const float*)d_in[4];
    const float*     b2   = (const float*)d_in[5];
    float*           out  = (float*)d_out;

    const long long* rows = ei;             // edge_index[0] : sources
    const long long* cols = ei + N_EDGES;   // edge_index[1] : targets

    // Workspace layout (~52 MB): dis | xw | h
    float* dis = (float*)d_ws;                       // [N]
    float* xw  = dis + N_NODES;                      // [N,64]
    float* h   = xw + (size_t)N_NODES * DFEAT;       // [N,64]

    const int THREADS = 256;
    const int n_waves_scatter = N_EDGES + N_NODES;              // 1.7M waves
    const int scatter_blocks  = (n_waves_scatter * 32 + THREADS - 1) / THREADS;
    const int n_tiles         = (N_NODES / 16) * (DFEAT / 16);  // 25000 waves
    const int gemm_blocks     = (n_tiles * 32 + THREADS - 1) / THREADS;
    const int node_blocks     = (N_NODES + THREADS - 1) / THREADS;
    const int edge_blocks     = (N_EDGES + THREADS - 1) / THREADS;
    const int feat_blocks     = (N_NODES * DFEAT + THREADS - 1) / THREADS;

    // deg^{-1/2} (self-loops included via init=1)
    gcn_deg_init <<<node_blocks, THREADS, 0, stream>>>(dis, N_NODES);
    gcn_deg_count<<<edge_blocks, THREADS, 0, stream>>>(cols, dis, N_EDGES);
    gcn_deg_rsqrt<<<node_blocks, THREADS, 0, stream>>>(dis, N_NODES);

    // Layer 1: xw = x @ W1 ; h = b1 + scatter(norm * xw[row] -> col)
    gcn_gemm_wmma<false><<<gemm_blocks, THREADS, 0, stream>>>(x, W1, xw);
    gcn_bias_init<<<feat_blocks, THREADS, 0, stream>>>(h, b1, N_NODES * DFEAT);
    gcn_scatter  <<<scatter_blocks, THREADS, 0, stream>>>(xw, rows, cols, dis, h,
                                                          N_EDGES, N_NODES);

    // Layer 2: xw = relu(h) @ W2 ; out = b2 + scatter(norm * xw[row] -> col)
    gcn_gemm_wmma<true><<<gemm_blocks, THREADS, 0, stream>>>(h, W2, xw);
    gcn_bias_init<<<feat_blocks, THREADS, 0, stream>>>(out, b2, N_NODES * DFEAT);
    gcn_scatter  <<<scatter_blocks, THREADS, 0, stream>>>(xw, rows, cols, dis, out,
                                                          N_EDGES, N_NODES);
}